// MassSpringSolver_11046655885740
// MI455X (gfx1250) — compile-verified
//
#include <hip/hip_runtime.h>

#define H2F   (0.01f * 0.01f)
#define KSF   1000.0f
#define CGIT  30
#define EPSF  1e-8f

typedef __attribute__((ext_vector_type(2))) float v2f;
typedef __attribute__((ext_vector_type(8))) float v8f;

// Exact wave32 sum using V_WMMA_F32_16X16X4_F32.
// A (16x4): VGPR0 lane L<16 -> A[L,0], lane L>=16 -> A[L-16,2]; VGPR1 (K=1,3) = 0.
// B = all ones  =>  D[m,n] = x_m + x_{m+16} for every n.
// Per lane, sum of the 8 accumulator VGPRs covers half the lanes; xor-16 add completes it.
// Must be called with EXEC all ones (fully reconverged, no early returns).
__device__ __forceinline__ float wave_sum_wmma(float x) {
  v2f a; a[0] = x;    a[1] = 0.0f;
  v2f b; b[0] = 1.0f; b[1] = 1.0f;
  v8f c = {};
  c = __builtin_amdgcn_wmma_f32_16x16x4_f32(false, a, false, b, (short)0, c, false, false);
  float s = ((c[0] + c[1]) + (c[2] + c[3])) + ((c[4] + c[5]) + (c[6] + c[7]));
  s += __shfl_xor(s, 16, 32);
  return s;
}

// Block (256 = 8 wave32) reduction -> single global atomic per block.
__device__ __forceinline__ void block_sum_atomic(float partial, float* __restrict__ dst) {
  __shared__ float lds[8];
  float ws = wave_sum_wmma(partial);              // EXEC all ones here
  if ((threadIdx.x & 31) == 0) lds[threadIdx.x >> 5] = ws;
  __syncthreads();
  if (threadIdx.x == 0) {
    float s = 0.0f;
#pragma unroll
    for (int w = 0; w < 8; ++w) s += lds[w];
    atomicAdd(dst, s);
  }
}

// ---- setup: compact per-edge spring data: Jc = (u.x, u.y, u.z, a) ----
// Jx*dv = a*dv - u*(u . dv),  a = KS*rl/l - KS,  u = (d/l)*sqrt(KS*rl/l)
__global__ void msk_jac(const float* __restrict__ pos, const int2* __restrict__ edges,
                        const float* __restrict__ rl, float4* __restrict__ Jc,
                        float* __restrict__ scal, int NE) {
  int e = blockIdx.x * blockDim.x + threadIdx.x;
  if (e < 8 && blockIdx.x == 0) scal[e] = 0.0f;   // one-time scalar zeroing
  if (e >= NE) return;
  int2 ed = edges[e];
  float dx = pos[3 * ed.x + 0] - pos[3 * ed.y + 0];
  float dy = pos[3 * ed.x + 1] - pos[3 * ed.y + 1];
  float dz = pos[3 * ed.x + 2] - pos[3 * ed.y + 2];
  float l   = sqrtf(dx * dx + dy * dy + dz * dz) + EPSF;
  float inv = 1.0f / l;
  float cc  = KSF * rl[e] * inv;       // KS*rl/l  (>= 0)
  float s   = sqrtf(cc) * inv;         // sqrt(c)/l
  Jc[e] = make_float4(dx * s, dy * s, dz * s, cc - KSF);
}

// x = 0, r = p = b, scal[0] += dot(b,b)    (4 elements / thread)
__global__ void msk_init(const float* __restrict__ rhs, float* __restrict__ x,
                         float* __restrict__ r, float* __restrict__ p,
                         float* __restrict__ scal, int n) {
  int base = (blockIdx.x * blockDim.x + threadIdx.x) * 4;
  float acc = 0.0f;
  if (base + 3 < n) {
    float4 v = *(const float4*)(rhs + base);
    *(float4*)(x + base) = make_float4(0.f, 0.f, 0.f, 0.f);
    *(float4*)(r + base) = v;
    *(float4*)(p + base) = v;
    acc = v.x * v.x + v.y * v.y + v.z * v.z + v.w * v.w;
  } else {
#pragma unroll
    for (int k = 0; k < 4; ++k) {
      int i = base + k;
      if (i < n) { float v = rhs[i]; x[i] = 0.f; r[i] = v; p[i] = v; acc += v * v; }
    }
  }
  block_sum_atomic(acc, &scal[0]);    // reconverged before WMMA
}

// Ap := 0 ; pAp := 0 ; rs_new := 0 ; (adv: rs <- rs_new of previous iteration)
__global__ void msk_clear(float* __restrict__ Ap, int n, float* __restrict__ scal, int adv) {
  int base = (blockIdx.x * blockDim.x + threadIdx.x) * 4;
  if (base + 3 < n) {
    *(float4*)(Ap + base) = make_float4(0.f, 0.f, 0.f, 0.f);
  } else {
#pragma unroll
    for (int k = 0; k < 4; ++k) if (base + k < n) Ap[base + k] = 0.0f;
  }
  if (base == 0) {
    if (adv) scal[0] = scal[2];       // rs <- rs_new (before zeroing rs_new)
    scal[1] = 0.0f;
    scal[2] = 0.0f;
  }
}

// scatter:  Jv[e0] += f, Jv[e1] -= f   with f = a*dv - u*(u.dv)
__global__ void msk_edge(const float4* __restrict__ Jc, const int2* __restrict__ edges,
                         const float* __restrict__ p, float* __restrict__ Jv, int NE) {
  int e = blockIdx.x * blockDim.x + threadIdx.x;
  if (e >= NE) return;
  // speculative prefetch of the streamed arrays (global_prefetch_b8)
  __builtin_prefetch((const char*)Jc    + (size_t)e * 16u + 16384u, 0, 1);
  __builtin_prefetch((const char*)edges + (size_t)e *  8u +  8192u, 0, 1);
  float4 j = Jc[e];
  int2  ed = edges[e];
  float dx = p[3 * ed.x + 0] - p[3 * ed.y + 0];
  float dy = p[3 * ed.x + 1] - p[3 * ed.y + 1];
  float dz = p[3 * ed.x + 2] - p[3 * ed.y + 2];
  float t  = j.x * dx + j.y * dy + j.z * dz;
  float fx = j.w * dx - j.x * t;
  float fy = j.w * dy - j.y * t;
  float fz = j.w * dz - j.z * t;
  atomicAdd(&Jv[3 * ed.x + 0], fx);  atomicAdd(&Jv[3 * ed.x + 1], fy);  atomicAdd(&Jv[3 * ed.x + 2], fz);
  atomicAdd(&Jv[3 * ed.y + 0], -fx); atomicAdd(&Jv[3 * ed.y + 1], -fy); atomicAdd(&Jv[3 * ed.y + 2], -fz);
}

// Ap = mass*p - h^2*Jv ; scal[1] += dot(p, Ap)
__global__ void msk_vertex(const float* __restrict__ p, const float* __restrict__ mass,
                           float* __restrict__ Ap, float* __restrict__ scal, int NV) {
  int i = blockIdx.x * blockDim.x + threadIdx.x;
  float dot = 0.0f;
  if (i < NV) {
    float m = mass[i];
#pragma unroll
    for (int k = 0; k < 3; ++k) {
      float pv = p[3 * i + k];
      float av = m * pv - H2F * Ap[3 * i + k];
      Ap[3 * i + k] = av;
      dot += pv * av;
    }
  }
  block_sum_atomic(dot, &scal[1]);
}

// alpha = rs/(pAp+eps); x += alpha*p; r -= alpha*Ap; scal[2] += dot(r,r)
__global__ void msk_xr(float* __restrict__ x, float* __restrict__ r,
                       const float* __restrict__ p, const float* __restrict__ Ap,
                       float* __restrict__ scal, int n) {
  int base = (blockIdx.x * blockDim.x + threadIdx.x) * 4;
  float alpha = scal[0] / (scal[1] + EPSF);
  float acc = 0.0f;
  if (base + 3 < n) {
    float4 pv = *(const float4*)(p  + base);
    float4 av = *(const float4*)(Ap + base);
    float4 xv = *(float4*)(x + base);
    float4 rv = *(float4*)(r + base);
    xv.x += alpha * pv.x; xv.y += alpha * pv.y; xv.z += alpha * pv.z; xv.w += alpha * pv.w;
    rv.x -= alpha * av.x; rv.y -= alpha * av.y; rv.z -= alpha * av.z; rv.w -= alpha * av.w;
    *(float4*)(x + base) = xv;
    *(float4*)(r + base) = rv;
    acc = rv.x * rv.x + rv.y * rv.y + rv.z * rv.z + rv.w * rv.w;
  } else {
#pragma unroll
    for (int k = 0; k < 4; ++k) {
      int i = base + k;
      if (i < n) {
        x[i] += alpha * p[i];
        float rn = r[i] - alpha * Ap[i];
        r[i] = rn;
        acc += rn * rn;
      }
    }
  }
  block_sum_atomic(acc, &scal[2]);
}

// beta = rs_new/(rs+eps); p = r + beta*p
__global__ void msk_p(float* __restrict__ p, const float* __restrict__ r,
                      const float* __restrict__ scal, int n) {
  int base = (blockIdx.x * blockDim.x + threadIdx.x) * 4;
  float beta = scal[2] / (scal[0] + EPSF);
  if (base + 3 < n) {
    float4 rv = *(const float4*)(r + base);
    float4 pv = *(float4*)(p + base);
    pv.x = rv.x + beta * pv.x; pv.y = rv.y + beta * pv.y;
    pv.z = rv.z + beta * pv.z; pv.w = rv.w + beta * pv.w;
    *(float4*)(p + base) = pv;
  } else {
#pragma unroll
    for (int k = 0; k < 4; ++k) {
      int i = base + k;
      if (i < n) p[i] = r[i] + beta * p[i];
    }
  }
}

extern "C" void kernel_launch(void* const* d_in, const int* in_sizes, int n_in,
                              void* d_out, int out_size, void* d_ws, size_t ws_size,
                              hipStream_t stream) {
  const float* rhs  = (const float*)d_in[0];
  const float* pos  = (const float*)d_in[1];
  const float* mass = (const float*)d_in[2];
  const float* rest = (const float*)d_in[3];
  const int2*  edgs = (const int2*) d_in[4];
  const int NV = in_sizes[2];          // mass: (NV,)
  const int NE = in_sizes[3];          // rest_len: (NE,)
  const int n  = NV * 3;

  float* x  = (float*)d_out;
  char*  ws = (char*)d_ws;
  float4* Jc = (float4*)ws;                 size_t off = (size_t)NE * 16u;
  float* r   = (float*)(ws + off);          off += (size_t)n * 4u;
  float* p   = (float*)(ws + off);          off += (size_t)n * 4u;
  float* Ap  = (float*)(ws + off);          off += (size_t)n * 4u;
  float* scal = (float*)(ws + off);         // [rs, pAp, rs_new, ...]

  const int B   = 256;                       // 8 wave32s
  const int gE  = (NE + B - 1) / B;
  const int gV  = (NV + B - 1) / B;
  const int gN4 = (n + 4 * B - 1) / (4 * B);

  msk_jac<<<gE, B, 0, stream>>>(pos, edgs, rest, Jc, scal, NE);
  msk_init<<<gN4, B, 0, stream>>>(rhs, x, r, p, scal, n);

  for (int it = 0; it < CGIT; ++it) {
    msk_clear<<<gN4, B, 0, stream>>>(Ap, n, scal, it);
    msk_edge<<<gE, B, 0, stream>>>(Jc, edgs, p, Ap, NE);
    msk_vertex<<<gV, B, 0, stream>>>(p, mass, Ap, scal, NV);
    msk_xr<<<gN4, B, 0, stream>>>(x, r, p, Ap, scal, n);
    msk_p<<<gN4, B, 0, stream>>>(p, r, scal, n);
  }
}